// MultiSimilarityLoss_75033078661970
// MI455X (gfx1250) — compile-verified
//
#include <hip/hip_runtime.h>
#include <math.h>

// ---------------------------------------------------------------------------
// Multi-Similarity loss, fused:  sim = feats @ feats^T  via V_WMMA_F32_16X16X4_F32
//  - A panel (32 rows x 1024) staged in LDS once per block via the Tensor Data
//    Mover (tensor_load_to_lds + s_wait_tensorcnt), with 1-DWORD/256-DWORD
//    padding so ds_load_b64 A fetches are bank-conflict-free (stride 1028).
//  - pass 1: per-row min-positive / max-negative (hard mining, LDS atomics)
//  - pass 2: masked exp sums -> log1p -> per-block partial; tiny reduce kernel.
// ---------------------------------------------------------------------------

typedef float v2f __attribute__((ext_vector_type(2)));
typedef float v8f __attribute__((ext_vector_type(8)));
typedef unsigned v4u __attribute__((ext_vector_type(4)));
typedef int v4i __attribute__((ext_vector_type(4)));
typedef int v8i __attribute__((ext_vector_type(8)));

#define B_DIM   4096
#define D_DIM   1024
#define M_TILE  32          // rows per workgroup (2 x 16-row WMMA subtiles)
#define THREADS 256         // 8 waves (wave32)
#define NWAVES  8
#define NTILES  32          // 4096 cols / (8 waves * 16 cols)
#define A_ROWSTRIDE 1028    // 1024 DWORDs + 4 pad DWORDs (TDM pad: 1 per 256)

#define SCALE_POS 2.0f
#define SCALE_NEG 40.0f
#define THRESH    0.5f
#define MARGIN    0.1f
#define ONE_MEPS  0.99999f  // 1 - 1e-5

// monotone float <-> uint mapping so integer LDS atomics implement float min/max
__device__ __forceinline__ unsigned f2mono(float f) {
  unsigned u = __float_as_uint(f);
  return (u & 0x80000000u) ? ~u : (u | 0x80000000u);
}
__device__ __forceinline__ float mono2f(unsigned m) {
  return __uint_as_float((m & 0x80000000u) ? (m & 0x7fffffffu) : ~m);
}

// 16x16 (x2 M-subtiles) x K=1024: A from LDS (padded layout), B from global/L2.
__device__ __forceinline__ void gemm_tiles(const float* __restrict__ a0p_base,
                                           const float* __restrict__ a1p_base,
                                           const float* __restrict__ bp,
                                           v8f& acc0, v8f& acc1) {
  for (int kb = 0; kb < 4; ++kb) {                 // 4 blocks of 256 DWORDs
    const float* a0p = a0p_base + kb * 257;        // +1 pad DWORD per block
    const float* a1p = a1p_base + kb * 257;
    const float* bkp = bp + kb * 256;
#pragma unroll 8
    for (int k2 = 0; k2 < 256; k2 += 4) {
      v2f a0 = *(const v2f*)(a0p + k2);            // ds_load_b64, conflict-free
      v2f a1 = *(const v2f*)(a1p + k2);
      v2f b  = *(const v2f*)(bkp + k2);            // global_load_b64 (L2-resident)
      acc0 = __builtin_amdgcn_wmma_f32_16x16x4_f32(false, a0, false, b,
                                                   (short)0, acc0, false, false);
      acc1 = __builtin_amdgcn_wmma_f32_16x16x4_f32(false, a1, false, b,
                                                   (short)0, acc1, false, false);
    }
  }
}

__global__ __launch_bounds__(THREADS)
void msloss_main_kernel(const float* __restrict__ feats,
                        const int*   __restrict__ labels,
                        float*       __restrict__ partials) {
  __shared__ float    A_sh[M_TILE * A_ROWSTRIDE];  // 131584 B, TDM-padded panel
  __shared__ int      lab_sh[B_DIM];               // 16 KB
  __shared__ unsigned minpos_u[M_TILE];
  __shared__ unsigned maxneg_u[M_TILE];
  __shared__ float    possum_sh[M_TILE];
  __shared__ float    negsum_sh[M_TILE];
  __shared__ float    rowloss_sh[M_TILE];

  const int tid  = threadIdx.x;
  const int wave = tid >> 5;
  const int lane = tid & 31;
  const int lo   = lane & 15;   // N (or M) index within half-wave
  const int hi   = lane >> 4;   // selects K pair / M+8 half per WMMA layout
  const int m0   = blockIdx.x * M_TILE;

  // ---- Tensor Data Mover: stage A panel (rows m0..m0+31, all K) into LDS ----
  if (wave == 0) {
    const unsigned lds_addr = (unsigned)(uintptr_t)(void*)A_sh;  // low 32b = LDS offset
    const unsigned long long ga =
        (unsigned long long)(uintptr_t)(feats + (size_t)m0 * D_DIM);
    v4u g0;
    g0.x = 1u;                                       // count=1 (valid user D#)
    g0.y = lds_addr;                                 // lds_addr [63:32]
    g0.z = (unsigned)(ga & 0xFFFFFFFFull);           // global_addr [95:64]
    g0.w = (unsigned)((ga >> 32) & 0x01FFFFFFull)    // global_addr [120:96]
         | (2u << 30);                               // type=2 ("image")
    v8i g1;
    g1[0] = (int)((2u << 16)      // data_size = 4 bytes
                | (1u << 20)      // pad_enable
                | (7u << 22));    // pad_interval = 256 DWORDs (pad_amount=0 -> 1 DW)
    g1[1] = (int)(1024u << 16);   // tensor_dim0[15:0] = 1024 (row length)
    g1[2] = (int)(4096u << 16);   // tensor_dim0 hi = 0; tensor_dim1[15:0] = 4096
    g1[3] = (int)(1024u << 16);   // tensor_dim1 hi = 0; tile_dim0 = 1024
    g1[4] = (int)(32u);           // tile_dim1 = 32 rows; tile_dim2 = 0
    g1[5] = (int)(1024u);         // tensor_dim0_stride low = 1024 elements
    g1[6] = 0;                    // stride hi / dim1_stride = 0
    g1[7] = 0;
    v4i g2 = {0, 0, 0, 0};
    v4i g3 = {0, 0, 0, 0};
#if __clang_major__ >= 23
    v8i g4 = {0, 0, 0, 0, 0, 0, 0, 0};
    __builtin_amdgcn_tensor_load_to_lds(g0, g1, g2, g3, g4, 0);
#else
    __builtin_amdgcn_tensor_load_to_lds(g0, g1, g2, g3, 0);
#endif
  }

  for (int i = tid; i < B_DIM; i += THREADS) lab_sh[i] = labels[i];
  if (tid < M_TILE) {
    minpos_u[tid]  = 0xFFFFFFFFu;  // mono(+NaN) -> min identity
    maxneg_u[tid]  = 0u;           // mono(-NaN) -> max identity
    possum_sh[tid] = 0.0f;
    negsum_sh[tid] = 0.0f;
  }
  if (wave == 0) __builtin_amdgcn_s_wait_tensorcnt(0);  // TDM done (wave 0)
  __syncthreads();                                      // publish to all waves

  // A operands from LDS: lane -> row lo (+16 for subtile 1), K pair 2*hi.
  // Padded row stride 1028 => bank index 4*lo+2*hi: conflict-free b64 loads.
  const float* a0p_base = A_sh + lo * A_ROWSTRIDE + 2 * hi;
  const float* a1p_base = a0p_base + 16 * A_ROWSTRIDE;

  // labels of the 8 rows this lane's acc VGPRs correspond to (M = v + 8*hi)
  int li0[8], li1[8];
#pragma unroll
  for (int v = 0; v < 8; ++v) {
    li0[v] = lab_sh[m0 + v + 8 * hi];
    li1[v] = lab_sh[m0 + 16 + v + 8 * hi];
  }

  // ---------------- pass 1: hard mining ----------------
  float minp0[8], minp1[8], maxn0[8], maxn1[8];
#pragma unroll
  for (int v = 0; v < 8; ++v) {
    minp0[v] = __builtin_inff();  minp1[v] = __builtin_inff();
    maxn0[v] = -__builtin_inff(); maxn1[v] = -__builtin_inff();
  }

  for (int t = 0; t < NTILES; ++t) {
    const int n0 = wave * 16 + t * (NWAVES * 16);
    const float* bp = feats + (size_t)(n0 + lo) * D_DIM + 2 * hi;
    v8f acc0 = {0.f,0.f,0.f,0.f,0.f,0.f,0.f,0.f};
    v8f acc1 = {0.f,0.f,0.f,0.f,0.f,0.f,0.f,0.f};
    gemm_tiles(a0p_base, a1p_base, bp, acc0, acc1);

    const int lj = lab_sh[n0 + lo];  // column label for this lane
#pragma unroll
    for (int v = 0; v < 8; ++v) {
      float s0 = acc0[v];
      float s1 = acc1[v];
      bool same0 = (li0[v] == lj);
      bool same1 = (li1[v] == lj);
      minp0[v] = (same0 && s0 < ONE_MEPS) ? fminf(minp0[v], s0) : minp0[v];
      minp1[v] = (same1 && s1 < ONE_MEPS) ? fminf(minp1[v], s1) : minp1[v];
      maxn0[v] = (!same0) ? fmaxf(maxn0[v], s0) : maxn0[v];
      maxn1[v] = (!same1) ? fmaxf(maxn1[v], s1) : maxn1[v];
    }
  }

  // merge per-lane results across lanes & waves via LDS integer atomics
#pragma unroll
  for (int v = 0; v < 8; ++v) {
    atomicMin(&minpos_u[v + 8 * hi],       f2mono(minp0[v]));
    atomicMin(&minpos_u[16 + v + 8 * hi],  f2mono(minp1[v]));
    atomicMax(&maxneg_u[v + 8 * hi],       f2mono(maxn0[v]));
    atomicMax(&maxneg_u[16 + v + 8 * hi],  f2mono(maxn1[v]));
  }
  __syncthreads();

  float mp0[8], mp1[8], mn0[8], mn1[8];
#pragma unroll
  for (int v = 0; v < 8; ++v) {
    mp0[v] = mono2f(minpos_u[v + 8 * hi]);
    mp1[v] = mono2f(minpos_u[16 + v + 8 * hi]);
    mn0[v] = mono2f(maxneg_u[v + 8 * hi]);
    mn1[v] = mono2f(maxneg_u[16 + v + 8 * hi]);
  }

  // ---------------- pass 2: masked exp sums ----------------
  float ps0[8], ps1[8], ns0[8], ns1[8];
#pragma unroll
  for (int v = 0; v < 8; ++v) { ps0[v]=0.f; ps1[v]=0.f; ns0[v]=0.f; ns1[v]=0.f; }

  for (int t = 0; t < NTILES; ++t) {
    const int n0 = wave * 16 + t * (NWAVES * 16);
    const float* bp = feats + (size_t)(n0 + lo) * D_DIM + 2 * hi;
    v8f acc0 = {0.f,0.f,0.f,0.f,0.f,0.f,0.f,0.f};
    v8f acc1 = {0.f,0.f,0.f,0.f,0.f,0.f,0.f,0.f};
    gemm_tiles(a0p_base, a1p_base, bp, acc0, acc1);

    const int lj = lab_sh[n0 + lo];
#pragma unroll
    for (int v = 0; v < 8; ++v) {
      float s0 = acc0[v];
      float s1 = acc1[v];
      bool same0 = (li0[v] == lj);
      bool same1 = (li1[v] == lj);

      bool psel0 = same0 && (s0 < ONE_MEPS) && (s0 < mn0[v] + MARGIN);
      bool psel1 = same1 && (s1 < ONE_MEPS) && (s1 < mn1[v] + MARGIN);
      bool nsel0 = (!same0) && (s0 > mp0[v] - MARGIN);
      bool nsel1 = (!same1) && (s1 > mp1[v] - MARGIN);

      float pe0 = __expf(-SCALE_POS * (s0 - THRESH));
      float pe1 = __expf(-SCALE_POS * (s1 - THRESH));
      float ne0 = __expf( SCALE_NEG * (s0 - THRESH));
      float ne1 = __expf( SCALE_NEG * (s1 - THRESH));

      ps0[v] += psel0 ? pe0 : 0.0f;
      ps1[v] += psel1 ? pe1 : 0.0f;
      ns0[v] += nsel0 ? ne0 : 0.0f;
      ns1[v] += nsel1 ? ne1 : 0.0f;
    }
  }

#pragma unroll
  for (int v = 0; v < 8; ++v) {
    atomicAdd(&possum_sh[v + 8 * hi],      ps0[v]);
    atomicAdd(&possum_sh[16 + v + 8 * hi], ps1[v]);
    atomicAdd(&negsum_sh[v + 8 * hi],      ns0[v]);
    atomicAdd(&negsum_sh[16 + v + 8 * hi], ns1[v]);
  }
  __syncthreads();

  if (tid < M_TILE) {
    float ps = possum_sh[tid];
    float ns = negsum_sh[tid];
    // sum>0 <=> any selected (smallest term e^-60 is a normal f32)
    bool valid = (ps > 0.0f) && (ns > 0.0f);
    rowloss_sh[tid] = valid
        ? (log1pf(ps) * (1.0f / SCALE_POS) + log1pf(ns) * (1.0f / SCALE_NEG))
        : 0.0f;
  }
  __syncthreads();

  if (tid == 0) {
    float s = 0.0f;
#pragma unroll
    for (int i = 0; i < M_TILE; ++i) s += rowloss_sh[i];
    partials[blockIdx.x] = s;
  }
}

__global__ void msloss_finalize_kernel(const float* __restrict__ partials,
                                       float* __restrict__ out, int n) {
  float v = 0.0f;
  for (int i = threadIdx.x; i < n; i += 32) v += partials[i];
#pragma unroll
  for (int off = 16; off > 0; off >>= 1) v += __shfl_down(v, off, 32);
  if (threadIdx.x == 0) out[0] = v / (float)B_DIM;
}

extern "C" void kernel_launch(void* const* d_in, const int* in_sizes, int n_in,
                              void* d_out, int out_size, void* d_ws, size_t ws_size,
                              hipStream_t stream) {
  const float* feats  = (const float*)d_in[0];
  const int*   labels = (const int*)d_in[1];  // jax default: int64 request -> int32
  float* out      = (float*)d_out;
  float* partials = (float*)d_ws;             // 128 floats

  const int nblocks = B_DIM / M_TILE;  // 128
  msloss_main_kernel<<<nblocks, THREADS, 0, stream>>>(feats, labels, partials);
  msloss_finalize_kernel<<<1, 32, 0, stream>>>(partials, out, nblocks);
}